// BNAFINN_51015621542177
// MI455X (gfx1250) — compile-verified
//
#include <hip/hip_runtime.h>
#include <math.h>

#define BATCH 4096
#define DIM   64
#define NH    128
#define HID   (DIM * NH)   // 8192
#define LDSP  132          // padded LDS row stride (floats), conflict-free

typedef __attribute__((ext_vector_type(2))) float v2f;
typedef __attribute__((ext_vector_type(8))) float v8f;

__device__ __forceinline__ v8f wmma4(v2f a, v2f b, v8f c) {
    // V_WMMA_F32_16X16X4_F32 : D = A(16x4) * B(4x16) + C(16x16), fp32
    return __builtin_amdgcn_wmma_f32_16x16x4_f32(false, a, false, b,
                                                 (short)0, c, false, false);
}

// fast, stable softplus: max(z,0) + log(1 + exp(-|z|)), hw transcendentals
__device__ __forceinline__ float sp_f(float z) {
    return fmaxf(z, 0.f) + __logf(1.f + __expf(-fabsf(z)));
}
// fast tanh via exp(-2|x|)
__device__ __forceinline__ float tanh_f(float x) {
    float e = __expf(-2.f * fabsf(x));
    float t = __fdividef(1.f - e, 1.f + e);
    return copysignf(t, x);
}

// ---------------------------------------------------------------------------
// prep1: weight-normalize W1 (8192x64, masked) and scatter into WMMA
// B-fragment order; also emit wpl1 diagonal term.
// fragment address for value B[k][n] = wn1[n][k]:
//   pos = (((n/16)*16 + k/4)*32 + (n%16) + 16*((k%4)/2))*2 + (k%2)
// ---------------------------------------------------------------------------
__global__ void bnaf_prep1(const float* __restrict__ W1,
                           const float* __restrict__ d1,
                           float* __restrict__ wn1p,
                           float* __restrict__ wpl1d) {
    int n = blockIdx.x * blockDim.x + threadIdx.x;
    if (n >= HID) return;
    int dc = n >> 7;                       // diagonal column = n / 128
    const float* row = W1 + (size_t)n * DIM;
    float acc = 0.f;
    for (int i = 0; i < DIM; ++i) {
        float Wv = row[i];
        float w  = (i == dc) ? expf(Wv) : ((i < dc) ? Wv : 0.f);
        acc += w * w;
    }
    float dn    = d1[n];
    float scale = expf(dn) * rsqrtf(acc);
    wpl1d[n]    = dn + row[dc] - 0.5f * logf(acc);
    int tn = n >> 4, lnb = n & 15;
    for (int i = 0; i < DIM; ++i) {
        float Wv = row[i];
        float w  = (i == dc) ? expf(Wv) : ((i < dc) ? Wv : 0.f);
        int kk   = i >> 2;
        int lane = lnb + (((i & 3) >> 1) << 4);
        int j    = i & 1;
        wn1p[(((tn * 16 + kk) * 32) + lane) * 2 + j] = scale * w;
    }
}

// ---------------------------------------------------------------------------
// prep2: weight-normalize W2 (64x8192, masked), scatter into fragment order,
// and build cmat[k] = wpl2diag(d, k) + wpl1diag[k]  (k = d*128 + k0).
// fragment address for B[k][d] = wn2[d][k]:
//   pos = (((k/128)*32 + (k/4)%32)*4 + d/16)*64 + ((d%16) + 16*((k%4)/2))*2 + (k%2)
// ---------------------------------------------------------------------------
__global__ void bnaf_prep2(const float* __restrict__ W2,
                           const float* __restrict__ d2,
                           const float* __restrict__ wpl1d,
                           float* __restrict__ wn2p,
                           float* __restrict__ cmat) {
    int d = threadIdx.x;
    if (d >= DIM) return;
    const float* row = W2 + (size_t)d * HID;
    int lo = d << 7, hi = lo + NH;
    float acc = 0.f;
    for (int k = 0; k < HID; ++k) {
        float Wv = row[k];
        float w  = (k >= lo && k < hi) ? expf(Wv) : ((k < lo) ? Wv : 0.f);
        acc += w * w;
    }
    float dn      = d2[d];
    float scale   = expf(dn) * rsqrtf(acc);
    float wplbase = dn - 0.5f * logf(acc);
    int td = d >> 4, lnb = d & 15;
    for (int k = 0; k < HID; ++k) {
        float Wv = row[k];
        float w  = (k >= lo && k < hi) ? expf(Wv) : ((k < lo) ? Wv : 0.f);
        int c    = k >> 7;
        int kk2  = (k >> 2) & 31;
        int lane = lnb + (((k & 3) >> 1) << 4);
        int j    = k & 1;
        wn2p[(((c * 32 + kk2) * 4 + td) * 64) + lane * 2 + j] = scale * w;
        if (k >= lo && k < hi) cmat[k] = wplbase + Wv + wpl1d[k];
    }
}

// ---------------------------------------------------------------------------
// Fused flow kernel: one wave per 16 batch rows.
//   preH = x @ wn1^T + b1  (WMMA f32 16x16x4; K-loop trip = ch/4+1, unroll 4)
//   h = tanh(preH); tg = -2*(preH - ln2 + softplus(-2 preH))
//   grad[b,d] = logsumexp_k( c[d,k] + tg[b, d*128+k] )   (d == chunk index)
//   y = h @ wn2^T + b2     (WMMA, A from LDS; per-tile uniform triangle guard)
//   gate mix + reversal + log-det accumulation
// ---------------------------------------------------------------------------
__global__ void __launch_bounds__(32)
bnaf_flow(const float* __restrict__ xin, float* __restrict__ xout,
          const float* __restrict__ wn1p, const float* __restrict__ b1,
          const float* __restrict__ wn2p, const float* __restrict__ b2,
          const float* __restrict__ cmat, const float* __restrict__ gates,
          float* __restrict__ ldws, float* __restrict__ ldout, int flow) {
    const int l    = threadIdx.x;
    const int lx   = l & 15;
    const int hl   = l >> 4;
    const int row0 = blockIdx.x * 16;

    __shared__ float h_lds[16 * LDSP];
    __shared__ float tg_lds[16 * LDSP];
    __shared__ float red[64];

    // A-fragment base for GEMM1: lane l covers row (l%16), cols 4kk+2*hl+{0,1}
    const float* xrA = xin + (size_t)(row0 + lx) * DIM + 2 * hl;

    const bool hasGate = (flow < 4);
    float gv = 0.f;
    if (hasGate) gv = gates[flow];
    const float sg  = 1.f / (1.f + __expf(-gv));
    const float spg = sp_f(gv);

    v8f yacc[4];
    #pragma unroll
    for (int td = 0; td < 4; ++td) yacc[td] = (v8f)0.f;

    float ldacc = 0.f;

    for (int ch = 0; ch < 64; ++ch) {
        const int n0  = ch * NH;
        const int kks = (ch >> 2) + 1;   // GEMM1 K-steps needed (triangular mask)

        // prefetch next chunk's weight fragments into caches (global_prefetch)
        if (ch < 63) {
            const char* p1 = (const char*)(wn1p + (size_t)((ch + 1) * 8 * 16) * 64);
            const char* p2 = (const char*)(wn2p + (size_t)(ch + 1) * 8192);
            #pragma unroll
            for (int i = 0; i < 8; ++i) {
                __builtin_prefetch(p1 + (i * 32 + l) * 128, 0, 1);
                __builtin_prefetch(p2 + (i * 32 + l) * 128, 0, 1);
            }
        }

        // ---- GEMM1 (8 tiles of 16 cols) + tanh + log-grad into LDS ----
        #pragma unroll 2
        for (int t = 0; t < 8; ++t) {
            float bv = b1[n0 + 16 * t + lx];
            v8f acc = {bv, bv, bv, bv, bv, bv, bv, bv};
            const float* bp = wn1p + (size_t)(((ch * 8 + t) * 16) * 32 + l) * 2;
            #pragma unroll 4
            for (int kk = 0; kk < kks; ++kk) {   // unrolled x4: loads batched
                v2f af = *(const v2f*)(xrA + 4 * kk);
                v2f bf = *(const v2f*)(bp + kk * 64);
                acc = wmma4(af, bf, acc);
            }
            float cv = cmat[n0 + 16 * t + lx];
            #pragma unroll
            for (int j = 0; j < 8; ++j) {
                float p  = acc[j];
                float tg = -2.f * (p - 0.6931471805599453f + sp_f(-2.f * p));
                int a    = (j + 8 * hl) * LDSP + 16 * t + lx;
                h_lds[a]  = tanh_f(p);
                tg_lds[a] = cv + tg;
            }
        }
        asm volatile("s_wait_dscnt 0" ::: "memory");

        // ---- logsumexp over the 128 cols of this chunk (== output dim ch) ----
        {
            float m = -INFINITY, s = 0.f;
            const float* tp = tg_lds + lx * LDSP + hl * 64;
            #pragma unroll 4
            for (int i = 0; i < 64; ++i) {   // branchless online LSE
                float v  = tp[i];
                float mn = fmaxf(m, v);
                s = s * __expf(m - mn) + __expf(v - mn);
                m = mn;
            }
            red[l]      = m;
            red[32 + l] = s;
            asm volatile("s_wait_dscnt 0" ::: "memory");
            if (l < 16) {
                float m2 = red[l + 16], s2 = red[48 + l];
                float M  = fmaxf(m, m2);
                float S  = s * __expf(m - M) + s2 * __expf(m2 - M);
                float lse = M + __logf(S);
                ldacc += hasGate ? (sp_f(lse + gv) - spg) : lse;
            }
        }

        // ---- GEMM2 accumulate: y(16x64) += h(16x128) @ wn2^T chunk ----
        {
            const int td0   = ch >> 4;       // y-tiles below the diagonal are zero
            const float* ap = h_lds + lx * LDSP + 2 * hl;
            #pragma unroll
            for (int td = 0; td < 4; ++td) { // compile-time td: yacc stays in regs
                if (td >= td0) {             // ONE uniform branch per tile
                    const float* bp = wn2p + (size_t)ch * 8192 + td * 64 + l * 2;
                    v8f acc = yacc[td];
                    #pragma unroll 4
                    for (int kk2 = 0; kk2 < 32; ++kk2) {
                        v2f af = *(const v2f*)(ap + 4 * kk2);
                        v2f bf = *(const v2f*)(bp + kk2 * 256);
                        acc = wmma4(af, bf, acc);
                    }
                    yacc[td] = acc;
                }
            }
        }
        asm volatile("s_wait_dscnt 0" ::: "memory"); // LDS reuse next chunk
    }

    // ---- epilogue: bias, gate mix, reversal, stores ----
    const float invsg = 1.f - sg;
    if (flow < 4) {
        #pragma unroll
        for (int t2 = 0; t2 < 4; ++t2) {
            #pragma unroll
            for (int j = 0; j < 8; ++j) {
                int r = j + 8 * hl, grow = row0 + r, col = 16 * t2 + lx;
                float yv = yacc[t2][j] + b2[col];
                float xv = xin[(size_t)grow * DIM + col];
                xout[(size_t)grow * DIM + (63 - col)] = sg * yv + invsg * xv;
            }
        }
        if (l < 16) {
            int grow = row0 + l;
            if (flow == 0) ldws[grow] = ldacc;
            else           ldws[grow] += ldacc;
        }
    } else {
        #pragma unroll
        for (int t2 = 0; t2 < 4; ++t2) {
            #pragma unroll
            for (int j = 0; j < 8; ++j) {
                int r = j + 8 * hl, grow = row0 + r, col = 16 * t2 + lx;
                xout[(size_t)grow * DIM + col] = yacc[t2][j] + b2[col];
            }
        }
        if (l < 16) {
            int grow = row0 + l;
            ldout[grow] = ldws[grow] + ldacc;
        }
    }
}

// ---------------------------------------------------------------------------
extern "C" void kernel_launch(void* const* d_in, const int* in_sizes, int n_in,
                              void* d_out, int out_size, void* d_ws, size_t ws_size,
                              hipStream_t stream) {
    const float* x     = (const float*)d_in[0];
    const float* W1    = (const float*)d_in[1];
    const float* d1    = (const float*)d_in[2];
    const float* b1    = (const float*)d_in[3];
    const float* W2    = (const float*)d_in[4];
    const float* d2    = (const float*)d_in[5];
    const float* b2    = (const float*)d_in[6];
    const float* gates = (const float*)d_in[7];

    float* ws    = (float*)d_ws;
    float* wn1p  = ws;                       // 524288
    float* wn2p  = wn1p + 524288;            // 524288
    float* cmat  = wn2p + 524288;            // 8192
    float* wpl1d = cmat + 8192;              // 8192
    float* xa    = wpl1d + 8192;             // 262144
    float* xb    = xa + BATCH * DIM;         // 262144
    float* ldws  = xb + BATCH * DIM;         // 4096

    float* outx  = (float*)d_out;
    float* outld = outx + (size_t)BATCH * DIM;

    const float* xcur = x;
    for (int f = 0; f < 5; ++f) {
        bnaf_prep1<<<HID / 256, 256, 0, stream>>>(W1 + (size_t)f * HID * DIM,
                                                  d1 + (size_t)f * HID,
                                                  wn1p, wpl1d);
        bnaf_prep2<<<1, 64, 0, stream>>>(W2 + (size_t)f * DIM * HID,
                                         d2 + (size_t)f * DIM,
                                         wpl1d, wn2p, cmat);
        float* xnext = (f == 4) ? outx : ((f & 1) ? xa : xb);
        float* ldo   = (f == 4) ? outld : ldws;
        bnaf_flow<<<BATCH / 16, 32, 0, stream>>>(xcur, xnext, wn1p,
                                                 b1 + (size_t)f * HID,
                                                 wn2p,
                                                 b2 + (size_t)f * DIM,
                                                 cmat, gates, ldws, ldo, f);
        xcur = xnext;
    }
}